// TSNE_loss_81570018885862
// MI455X (gfx1250) — compile-verified
//
#include <hip/hip_runtime.h>
#include <math.h>

typedef __attribute__((ext_vector_type(2))) float v2f;
typedef __attribute__((ext_vector_type(8))) float v8f;

#define TSNE_SCALE 3.5f
#define COL_SPLIT 8          // column chunks; all 8 waves of a block share one chunk
#define WAVES_PER_BLOCK 8
#define BLOCK_THREADS 256
#define N_FIXED 32768
#define SPAN (N_FIXED / COL_SPLIT)   // 4096 column points staged in LDS per block

__device__ __forceinline__ float wave_reduce_add(float v) {
#pragma unroll
  for (int off = 16; off > 0; off >>= 1)
    v += __shfl_down(v, off, 32);
  return v;
}

// d_ws accumulators: [0] = sum p*(log p + log1p(d2)), [1] = sum p, [2] = T (pairwise sum incl. diagonal)
__global__ void tsne_init_kernel(float* __restrict__ acc) {
  if (threadIdx.x < 3) acc[threadIdx.x] = 0.0f;
}

__global__ void __launch_bounds__(BLOCK_THREADS) tsne_edge_kernel(
    const float2* __restrict__ emb, const float* __restrict__ p,
    const int* __restrict__ heads, const int* __restrict__ tails,
    int E, float* __restrict__ acc) {
  float s1 = 0.0f, sp = 0.0f;
  int idx = blockIdx.x * blockDim.x + threadIdx.x;
  int stride = gridDim.x * blockDim.x;
  for (int e = idx; e < E; e += stride) {
    int h = heads[e];
    int t = tails[e];
    float2 eh = emb[h];
    float2 et = emb[t];
    float dx = (eh.x - et.x) * TSNE_SCALE;
    float dy = (eh.y - et.y) * TSNE_SCALE;
    float d2 = __fmaf_rn(dx, dx, dy * dy);
    float pe = p[e];
    s1 += pe * (__logf(pe) + log1pf(d2));
    sp += pe;
  }
  s1 = wave_reduce_add(s1);
  sp = wave_reduce_add(sp);
  if ((threadIdx.x & 31) == 0) {
    atomicAdd(&acc[0], s1);
    atomicAdd(&acc[1], sp);
  }
}

// Pairwise normalizer: one block = 8 row-strips x one shared column chunk.
// Column chunk is DMA'd global->LDS (ASYNCcnt path) and transformed in LDS into the
// WMMA B-operand layout:  sB[0..SPAN)   = (-2x_j, -2y_j)   (K=0,1 rows, lanes 0-15)
//                         sB[SPAN..2S)  = ( 1,  |e_j|^2 )  (K=2,3 rows, lanes 16-31)
// Hot loop (B software-pipelined): ds_load_b64 (next tile) + v_wmma_f32_16x16x4_f32
// (current tile) + pk_add/max/rcp accumulate.
__global__ void __launch_bounds__(BLOCK_THREADS) tsne_pair_kernel(
    const float2* __restrict__ emb, float* __restrict__ accT, int n) {
  __shared__ float2 sB[2 * SPAN];   // 64 KB

  const int lane = threadIdx.x & 31;
  const int wave = threadIdx.x >> 5;
  const int chunk = blockIdx.x % COL_SPLIT;
  const int stripGroup = blockIdx.x / COL_SPLIT;
  const int row0 = (stripGroup * WAVES_PER_BLOCK + wave) * 16;
  const int colBeg = chunk * SPAN;

  // A-operand row data: issue the global load early so it overlaps the async DMA.
  const int m = lane & 15;
  const bool hiLane = lane >= 16;
  float2 er = emb[row0 + m];

  // ---- Stage raw column chunk into LDS with async global->LDS loads ----
  {
    const int per = SPAN / BLOCK_THREADS;     // 16 points per thread
    const int i0 = threadIdx.x * per;
#pragma unroll
    for (int k = 0; k < per; ++k) {
      const int i = i0 + k;
      unsigned ldsoff = (unsigned)(uintptr_t)&sB[i];        // logical LDS byte address
      const float2* g = &emb[colBeg + i];
      asm volatile("global_load_async_to_lds_b64 %0, %1, off"
                   :: "v"(ldsoff), "v"(g) : "memory");
    }
    asm volatile("s_wait_asynccnt 0x0" ::: "memory");
    // In-place transform (each thread touches only the points it loaded)
#pragma unroll
    for (int k = 0; k < per; ++k) {
      const int i = i0 + k;
      float2 r = sB[i];
      const float sx = r.x * TSNE_SCALE;
      const float sy = r.y * TSNE_SCALE;
      float2 lo; lo.x = -2.0f * sx; lo.y = -2.0f * sy;
      float2 hi; hi.x = 1.0f;       hi.y = __fmaf_rn(sx, sx, sy * sy);
      sB[i] = lo;
      sB[SPAN + i] = hi;
    }
  }
  __syncthreads();

  // ---- A operand: lanes 0-15 hold [x_i, y_i] (K=0,1); lanes 16-31 hold [|e_i|^2, 1] (K=2,3)
  const float xr = er.x * TSNE_SCALE;
  const float yr = er.y * TSNE_SCALE;
  const float sqr = __fmaf_rn(xr, xr, yr * yr);
  v2f A;
  A.x = hiLane ? sqr : xr;
  A.y = hiLane ? 1.0f : yr;

  const int bbase = (hiLane ? SPAN : 0) + m;   // per-lane base into LDS B table

  // Software-pipelined B operand: prefetch tile c+16 while WMMA consumes tile c.
  v2f bCur;
  {
    float2 t = sB[bbase];
    bCur.x = t.x; bCur.y = t.y;
  }

  v8f acc = {0.f, 0.f, 0.f, 0.f, 0.f, 0.f, 0.f, 0.f};
#pragma unroll 2
  for (int c = 0; c < SPAN; c += 16) {
    const int cn = (c + 16) & (SPAN - 1);      // uniform wrap (SALU), no branch
    float2 t = sB[bbase + cn];                 // ds_load_b64 for NEXT tile
    v2f bNext; bNext.x = t.x; bNext.y = t.y;

    v8f d = {0.f, 0.f, 0.f, 0.f, 0.f, 0.f, 0.f, 0.f};
    d = __builtin_amdgcn_wmma_f32_16x16x4_f32(
        /*neg_a=*/false, A, /*neg_b=*/false, bCur,
        /*c_mod=*/(short)0, d, /*reuse_a=*/false, /*reuse_b=*/false);

#pragma unroll
    for (int r = 0; r < 8; ++r) {
      // 1 + max(d2,0) == max(1+d2, 1): v_pk_add_f32 pairs + single v_max_num_f32
      float den = fmaxf(d[r] + 1.0f, 1.0f);
      acc[r] += __builtin_amdgcn_rcpf(den);    // v_rcp_f32 (TRANS, co-executes with WMMA)
    }

    bCur = bNext;
  }

  float s = (acc[0] + acc[1]) + (acc[2] + acc[3]) +
            (acc[4] + acc[5]) + (acc[6] + acc[7]);
  s = wave_reduce_add(s);
  if (lane == 0) atomicAdd(accT, s);
}

__global__ void tsne_finalize_kernel(const float* __restrict__ acc,
                                     float* __restrict__ out, int n) {
  if (threadIdx.x == 0 && blockIdx.x == 0) {
    float T = acc[2];
    float Z = T - (float)n;     // remove diagonal (each diag similarity == 1)
    out[0] = acc[0] + acc[1] * __logf(Z);
  }
}

extern "C" void kernel_launch(void* const* d_in, const int* in_sizes, int n_in,
                              void* d_out, int out_size, void* d_ws, size_t ws_size,
                              hipStream_t stream) {
  const float2* emb   = (const float2*)d_in[0];  // [N,2] f32
  const float*  p     = (const float*)d_in[1];   // [E]   f32
  const int*    heads = (const int*)d_in[2];     // [E]   i32
  const int*    tails = (const int*)d_in[3];     // [E]   i32
  float* out = (float*)d_out;
  float* acc = (float*)d_ws;                     // 3 floats of scratch

  const int n = in_sizes[0] / 2;                 // 32768
  const int E = in_sizes[1];                     // ~3.1M

  tsne_init_kernel<<<1, 32, 0, stream>>>(acc);

  tsne_edge_kernel<<<1024, BLOCK_THREADS, 0, stream>>>(emb, p, heads, tails, E, acc);

  const int strips = n / 16;                                    // 2048
  const int blocks = (strips / WAVES_PER_BLOCK) * COL_SPLIT;    // 2048
  tsne_pair_kernel<<<blocks, BLOCK_THREADS, 0, stream>>>(emb, acc + 2, n);

  tsne_finalize_kernel<<<1, 32, 0, stream>>>(acc, out, n);
}